// Separateq0qgqkckGNNReachabilityNet_34316788695694
// MI455X (gfx1250) — compile-verified
//
#include <hip/hip_runtime.h>
#include <hip/hip_bf16.h>
#include <math.h>

#define BN 32
#define NN 2048
#define NE 65536

typedef float v2f __attribute__((ext_vector_type(2)));
typedef float v8f __attribute__((ext_vector_type(8)));

__device__ __forceinline__ float lrelu(float x) { return x > 0.f ? x : 0.01f * x; }

__device__ __forceinline__ v8f wmma4(v2f a, v2f b, v8f c) {
    // D = A(16x4) x B(4x16) + C(16x16), exact fp32
    return __builtin_amdgcn_wmma_f32_16x16x4_f32(
        /*neg_a=*/false, a, /*neg_b=*/false, b,
        /*c_mod=*/(short)0, c, /*reuse_a=*/false, /*reuse_b=*/false);
}

// ---------------------------------------------------------------------------
// Kernel 1: per-node vertex MLP  -> vf[b][n][34]  (32 features + col 2)
// ---------------------------------------------------------------------------
__global__ void __launch_bounds__(256) vertex_kernel(
    const float* __restrict__ vertices,
    const float* __restrict__ Wx,  const float* __restrict__ bx,
    const float* __restrict__ Wy,  const float* __restrict__ by,
    const float* __restrict__ Wth, const float* __restrict__ bth,
    const float* __restrict__ W1c, const float* __restrict__ b1c,
    const float* __restrict__ W2c, const float* __restrict__ b2c,
    const float* __restrict__ W1v, const float* __restrict__ b1v,
    const float* __restrict__ W2v, const float* __restrict__ b2v,
    float* __restrict__ vf)
{
    __shared__ float sWx[16], sWy[16], sWth[16], sbx[8], sby[8], sbth[8];
    __shared__ float sW1c[32 * 24], sb1c[32], sW2c[32 * 32], sb2c[32];
    __shared__ float sW1v[32 * 64], sb1v[32], sW2v[32 * 32], sb2v[32];

    const int tid = threadIdx.x;
    if (tid < 16) { sWx[tid] = Wx[tid]; sWy[tid] = Wy[tid]; sWth[tid] = Wth[tid]; }
    if (tid < 8)  { sbx[tid] = bx[tid]; sby[tid] = by[tid]; sbth[tid] = bth[tid]; }
    for (int i = tid; i < 32 * 24; i += 256) sW1c[i] = W1c[i];
    for (int i = tid; i < 32 * 32; i += 256) { sW2c[i] = W2c[i]; sW2v[i] = W2v[i]; }
    for (int i = tid; i < 32 * 64; i += 256) sW1v[i] = W1v[i];
    if (tid < 32) { sb1c[tid] = b1c[tid]; sb2c[tid] = b2c[tid];
                    sb1v[tid] = b1v[tid]; sb2v[tid] = b2v[tid]; }
    __syncthreads();

    const int n = blockIdx.x * 256 + tid;
    const int b = blockIdx.y;
    const float* vrow = vertices + ((size_t)b * NN + n) * 11;

    float qk[3], q0[3], qg[3];
    #pragma unroll
    for (int i = 0; i < 3; ++i) { qk[i] = vrow[i]; q0[i] = vrow[3 + i]; qg[i] = vrow[6 + i]; }
    const float col0 = vrow[9], col1 = vrow[10];

    float cf[64];
    #pragma unroll
    for (int pass = 0; pass < 2; ++pass) {
        const float* qb = pass ? qg : q0;
        float c24[24];
        #pragma unroll
        for (int j = 0; j < 8; ++j) {
            c24[j]      = lrelu(sWx[2 * j] * qk[0] + sWx[2 * j + 1] * qb[0] + sbx[j]);
            c24[8 + j]  = lrelu(sWy[2 * j] * qk[1] + sWy[2 * j + 1] * qb[1] + sby[j]);
            c24[16 + j] = lrelu(sWth[2 * j] * qk[2] + sWth[2 * j + 1] * qb[2] + sbth[j]);
        }
        float h[32];
        #pragma unroll
        for (int j = 0; j < 32; ++j) {
            float t = sb1c[j];
            #pragma unroll
            for (int i = 0; i < 24; ++i) t += sW1c[j * 24 + i] * c24[i];
            h[j] = lrelu(t);
        }
        #pragma unroll
        for (int j = 0; j < 32; ++j) {
            float t = sb2c[j];
            #pragma unroll
            for (int i = 0; i < 32; ++i) t += sW2c[j * 32 + i] * h[i];
            cf[pass * 32 + j] = lrelu(t);
        }
    }
    float v1[32];
    #pragma unroll
    for (int j = 0; j < 32; ++j) {
        float t = sb1v[j];
        #pragma unroll
        for (int i = 0; i < 64; ++i) t += sW1v[j * 64 + i] * cf[i];
        v1[j] = lrelu(t);
    }
    float* orow = vf + ((size_t)b * NN + n) * 34;
    #pragma unroll
    for (int j = 0; j < 32; ++j) {
        float t = sb2v[j];
        #pragma unroll
        for (int i = 0; i < 32; ++i) t += sW2v[j * 32 + i] * v1[i];
        orow[j] = lrelu(t);
    }
    orow[32] = col0;
    orow[33] = col1;
}

// ---------------------------------------------------------------------------
// Kernel 2: per-dst in-degree (shared across batches)
// ---------------------------------------------------------------------------
__global__ void cnt_kernel(const int* __restrict__ edges, float* __restrict__ cnt) {
    const int e = blockIdx.x * 256 + threadIdx.x;
    atomicAdd(&cnt[edges[NE + e]], 1.0f);
}

// ---------------------------------------------------------------------------
// Kernel 3: edge MLP (WMMA f32) + scatter-add.  One wave per 16 edges.
//   GEMM1: pair(16x68) @ W1e.T(68x32), relu    -> 2x17 v_wmma_f32_16x16x4_f32
//   GEMM2: m(16x32)    @ W2e.T(32x32), lrelu   -> 2x8  v_wmma_f32_16x16x4_f32
// ---------------------------------------------------------------------------
__global__ void __launch_bounds__(256) edge_kernel(
    const int* __restrict__ edges, const float* __restrict__ vf,
    const float* __restrict__ W1e, const float* __restrict__ b1e,
    const float* __restrict__ W2e, const float* __restrict__ b2e,
    float* __restrict__ sums)
{
    __shared__ __align__(16) float sW1e[32 * 68];     // stride 68
    __shared__ __align__(16) float sW2e[32 * 34];     // padded stride 34 (bank-safe)
    __shared__ float sb1e[32], sb2e[32];
    __shared__ __align__(16) float pairbuf[8][16 * 68]; // per-wave; reused as m-tile
    __shared__ int sdst[8][16];

    const int tid  = threadIdx.x;
    const int wave = tid >> 5;
    const int lane = tid & 31;
    for (int i = tid; i < 32 * 68; i += 256) sW1e[i] = W1e[i];
    for (int i = tid; i < 32 * 32; i += 256) sW2e[(i >> 5) * 34 + (i & 31)] = W2e[i];
    if (tid < 32) { sb1e[tid] = b1e[tid]; sb2e[tid] = b2e[tid]; }
    __syncthreads();

    const int b  = blockIdx.y;
    const int e0 = (blockIdx.x * 8 + wave) * 16;
    const int h  = lane >> 4;   // lane half: src(0)/dst(1) loader; K-half in fragments
    const int c  = lane & 15;   // edge-row for staging; N-col / M-row for compute

    // ---- stage pair tile: row e = [vf[src[e]](34) | vf[dst[e]](34)] ----
    const int node = edges[h * NE + (e0 + c)];
    if (h) sdst[wave][c] = node;
    {
        const float* vrow = vf + ((size_t)b * NN + node) * 34;
        float* prow = &pairbuf[wave][c * 68 + h * 34];
        #pragma unroll
        for (int i = 0; i < 34; i += 2)
            *(float2*)(prow + i) = *(const float2*)(vrow + i);
    }
    // same-wave LDS ops retire in order: no barrier needed for wave-private tile

    // ---- GEMM1: K = 68 in 17 chunks of 4 ----
    v8f acc0 = {0,0,0,0,0,0,0,0}, acc1 = {0,0,0,0,0,0,0,0};
    const float* P = &pairbuf[wave][0];
    #pragma unroll
    for (int k = 0; k < 17; ++k) {
        const int kb = k * 4 + h * 2;                 // A/B lane-half K split
        v2f a, w0, w1;
        a[0]  = P[c * 68 + kb];          a[1]  = P[c * 68 + kb + 1];
        w0[0] = sW1e[c * 68 + kb];       w0[1] = sW1e[c * 68 + kb + 1];
        w1[0] = sW1e[(c + 16) * 68 + kb]; w1[1] = sW1e[(c + 16) * 68 + kb + 1];
        acc0 = wmma4(a, w0, acc0);
        acc1 = wmma4(a, w1, acc1);
    }
    // bias + relu; C-layout: lane holds col N=c, rows M=v+8*h
    const float bA = sb1e[c], bB = sb1e[c + 16];
    float* Mt = &pairbuf[wave][0];                    // pair data dead: reuse, stride 34
    #pragma unroll
    for (int v = 0; v < 8; ++v) {
        Mt[(v + 8 * h) * 34 + c]      = fmaxf(acc0[v] + bA, 0.0f);
        Mt[(v + 8 * h) * 34 + 16 + c] = fmaxf(acc1[v] + bB, 0.0f);
    }

    // ---- GEMM2: K = 32 in 8 chunks of 4 ----
    v8f d0 = {0,0,0,0,0,0,0,0}, d1 = {0,0,0,0,0,0,0,0};
    #pragma unroll
    for (int k = 0; k < 8; ++k) {
        const int kb = k * 4 + h * 2;
        v2f a, w0, w1;
        a[0]  = Mt[c * 34 + kb];          a[1]  = Mt[c * 34 + kb + 1];
        w0[0] = sW2e[c * 34 + kb];        w0[1] = sW2e[c * 34 + kb + 1];
        w1[0] = sW2e[(c + 16) * 34 + kb]; w1[1] = sW2e[(c + 16) * 34 + kb + 1];
        d0 = wmma4(a, w0, d0);
        d1 = wmma4(a, w1, d1);
    }

    // ---- bias + lrelu + scatter-add into sums[b][dst][0..31] ----
    const float c0 = sb2e[c], c1 = sb2e[c + 16];
    float* srow = sums + (size_t)b * NN * 32;
    #pragma unroll
    for (int v = 0; v < 8; ++v) {
        const int er = v + 8 * h;
        const int d  = sdst[wave][er];
        float x0 = d0[v] + c0; x0 = x0 > 0.f ? x0 : 0.01f * x0;
        float x1 = d1[v] + c1; x1 = x1 > 0.f ? x1 : 0.01f * x1;
        atomicAdd(srow + (size_t)d * 32 + c,      x0);
        atomicAdd(srow + (size_t)d * 32 + 16 + c, x1);
    }
}

// ---------------------------------------------------------------------------
// Kernel 4: agg = sums/max(cnt,1) -> output MLP -> per-batch dot with Wg
// ---------------------------------------------------------------------------
__global__ void __launch_bounds__(256) out_kernel(
    const float* __restrict__ sums, const float* __restrict__ cnt,
    const float* __restrict__ W1o, const float* __restrict__ b1o,
    const float* __restrict__ W2o, const float* __restrict__ b2o,
    const float* __restrict__ Wg, float* __restrict__ acc)
{
    __shared__ float sW1o[32 * 32], sb1o[32], sW2o[32], sb2o;
    __shared__ float red[256];
    const int tid = threadIdx.x;
    for (int i = tid; i < 32 * 32; i += 256) sW1o[i] = W1o[i];
    if (tid < 32) { sb1o[tid] = b1o[tid]; sW2o[tid] = W2o[tid]; }
    if (tid == 0) sb2o = b2o[0];
    __syncthreads();

    const int n = blockIdx.x * 256 + tid;
    const int b = blockIdx.y;
    const float* srow = sums + ((size_t)b * NN + n) * 32;
    const float inv = 1.0f / fmaxf(cnt[n], 1.0f);
    float a[32];
    #pragma unroll
    for (int i = 0; i < 32; ++i) a[i] = srow[i] * inv;
    float s = sb2o;
    #pragma unroll
    for (int j = 0; j < 32; ++j) {
        float t = sb1o[j];
        #pragma unroll
        for (int i = 0; i < 32; ++i) t += sW1o[j * 32 + i] * a[i];
        s += sW2o[j] * fmaxf(t, 0.0f);
    }
    red[tid] = lrelu(s) * Wg[n];
    __syncthreads();
    #pragma unroll
    for (int st = 128; st > 0; st >>= 1) {
        if (tid < st) red[tid] += red[tid + st];
        __syncthreads();
    }
    if (tid == 0) atomicAdd(&acc[b], red[0]);
}

__global__ void fin_kernel(const float* __restrict__ acc, const float* __restrict__ bg,
                           float* __restrict__ out) {
    const int b = threadIdx.x;
    if (b < BN) out[b] = 1.0f / (1.0f + __expf(-(acc[b] + bg[0])));
}

// ---------------------------------------------------------------------------
extern "C" void kernel_launch(void* const* d_in, const int* in_sizes, int n_in,
                              void* d_out, int out_size, void* d_ws, size_t ws_size,
                              hipStream_t stream) {
    const float* vertices = (const float*)d_in[0];
    const int*   edges    = (const int*)d_in[1];
    const float* Wx  = (const float*)d_in[2],  *bx  = (const float*)d_in[3];
    const float* Wy  = (const float*)d_in[4],  *by  = (const float*)d_in[5];
    const float* Wth = (const float*)d_in[6],  *bth = (const float*)d_in[7];
    const float* W1c = (const float*)d_in[8],  *b1c = (const float*)d_in[9];
    const float* W2c = (const float*)d_in[10], *b2c = (const float*)d_in[11];
    const float* W1v = (const float*)d_in[12], *b1v = (const float*)d_in[13];
    const float* W2v = (const float*)d_in[14], *b2v = (const float*)d_in[15];
    const float* W1e = (const float*)d_in[16], *b1e = (const float*)d_in[17];
    const float* W2e = (const float*)d_in[18], *b2e = (const float*)d_in[19];
    const float* W1o = (const float*)d_in[20], *b1o = (const float*)d_in[21];
    const float* W2o = (const float*)d_in[22], *b2o = (const float*)d_in[23];
    const float* Wg  = (const float*)d_in[24], *bg  = (const float*)d_in[25];

    float* ws = (float*)d_ws;
    const size_t off_vf   = 0;                                  // B*N*34
    const size_t off_sums = off_vf + (size_t)BN * NN * 34;      // B*N*32
    const size_t off_cnt  = off_sums + (size_t)BN * NN * 32;    // N
    const size_t off_acc  = off_cnt + NN;                       // B

    hipMemsetAsync(ws + off_sums, 0,
                   ((size_t)BN * NN * 32 + NN + BN) * sizeof(float), stream);

    vertex_kernel<<<dim3(NN / 256, BN), 256, 0, stream>>>(
        vertices, Wx, bx, Wy, by, Wth, bth,
        W1c, b1c, W2c, b2c, W1v, b1v, W2v, b2v, ws + off_vf);

    cnt_kernel<<<NE / 256, 256, 0, stream>>>(edges, ws + off_cnt);

    edge_kernel<<<dim3(NE / 128, BN), 256, 0, stream>>>(
        edges, ws + off_vf, W1e, b1e, W2e, b2e, ws + off_sums);

    out_kernel<<<dim3(NN / 256, BN), 256, 0, stream>>>(
        ws + off_sums, ws + off_cnt, W1o, b1o, W2o, b2o, Wg, ws + off_acc);

    fin_kernel<<<1, 32, 0, stream>>>(ws + off_acc, bg, (float*)d_out);
}